// GaussianVectorQuantizer_27504970563601
// MI455X (gfx1250) — compile-verified
//
#include <hip/hip_runtime.h>

typedef __attribute__((ext_vector_type(2))) float v2f;
typedef __attribute__((ext_vector_type(8))) float v8f;

// ---------------- problem constants ----------------
#define NK        512          // codebook size
#define NDIM      64           // code dim
#define NTOT      32768        // bs*W*H
#define NBS       32
#define ROWS_WG   64           // rows per workgroup
#define NWAVES    4            // 128 threads
#define NWG       (NTOT / ROWS_WG)

// ---------------- LDS layout (floats) ----------------
#define CB_PITCH  68                               // 512x68 codebook (padded, 16B-aligned rows)
#define CB_OFF    0
#define LG_OFF    (CB_OFF + NK * CB_PITCH)         // 64x512 raw logits (later: exp values)
#define ZT_OFF    (LG_OFF + ROWS_WG * NK)          // 64x68  z transposed [d][r]
#define ZQ_OFF    (ZT_OFF + NDIM * 68)             // 64x68  zq transposed [d][r]
#define CN_OFF    (ZQ_OFF + NDIM * 68)             // 512 codebook norms
#define AP_OFF    (CN_OFF + NK)                    // 512 per-code prob accum
#define WB_OFF    (AP_OFF + NK)                    // 4 waves x 16x20 weight bounce
#define SC_OFF    (WB_OFF + NWAVES * 16 * 20)      // 16 scratch
#define LDS_FLOATS (SC_OFF + 16)

__device__ __forceinline__ v8f wmma_f32(v2f a, v2f b, v8f c) {
  // D = A(16x4,f32) x B(4x16,f32) + C(16x16,f32)
  return __builtin_amdgcn_wmma_f32_16x16x4_f32(false, a, false, b, (short)0, c,
                                               false, false);
}

__global__ void __launch_bounds__(128, 1)
gvq_main(const float* __restrict__ z, const float* __restrict__ var_q,
         const float* __restrict__ cb, const float* __restrict__ gu,
         float* __restrict__ out, float* __restrict__ ws) {
  extern __shared__ float sm[];
  const int tid   = threadIdx.x;
  const int lane  = tid & 31;
  const int wave  = tid >> 5;
  const int rlane = lane & 15;       // 0..15
  const int half  = lane >> 4;       // 0 or 1
  const int wg    = blockIdx.x;

  // precision (clamped as in reference)
  const float vq   = fmaxf(var_q[0], 1e-5f);
  const float hp   = 0.5f * fminf(1.0f / vq, 1e5f);
  const float hp2  = 2.0f * hp;

  // ---- pass 0a: codebook -> LDS (padded), zero avg-prob accum ----
  for (int i = tid; i < NK * NDIM / 4; i += 128) {
    const float4 v = ((const float4*)cb)[i];
    const int c = i >> 4, d = (i & 15) << 2;
    *(float4*)&sm[CB_OFF + c * CB_PITCH + d] = v;
  }
  for (int i = tid; i < NK; i += 128) sm[AP_OFF + i] = 0.0f;

  // ---- pass 0b: load this wave's 16 rows of z as WMMA A-fragments ----
  // A(16x4): lanes 0-15 row M=lane, k={0,1}; lanes 16-31 row M=lane-16, k={2,3}
  const int rowLocal = wave * 16 + rlane;
  const int nGlob    = wg * ROWS_WG + rowLocal;
  const int bb       = nGlob >> 10;        // / (W*H)
  const int rem      = nGlob & 1023;
  const size_t zbase = (size_t)bb * 65536 + rem;   // z[b, d, w, h]
  v2f  az[16];
  float znPart = 0.0f;
#pragma unroll
  for (int kk = 0; kk < 16; ++kk) {
    const int d0 = kk * 4 + half * 2;
    const float x0 = z[zbase + (size_t)d0 * 1024];
    const float x1 = z[zbase + (size_t)(d0 + 1) * 1024];
    az[kk].x = x0; az[kk].y = x1;
    znPart += x0 * x0 + x1 * x1;
    sm[ZT_OFF + d0 * 68 + rowLocal]       = x0;   // z transposed tile [d][r]
    sm[ZT_OFF + (d0 + 1) * 68 + rowLocal] = x1;
  }
  // full row norm: combine the two half-lane partials, then place into the
  // C/D accumulator row layout: stat[j] <-> row (j + 8*half)
  const float znFull = znPart + __shfl_xor(znPart, 16, 32);  // lane L: ||z_row(L%16)||^2
  float hzn[8];
#pragma unroll
  for (int j = 0; j < 8; ++j) hzn[j] = hp * __shfl(znFull, j + half * 8, 32);

  __syncthreads();

  // ---- codebook norms ----
  for (int c = tid; c < NK; c += 128) {
    float s = 0.0f;
    for (int d = 0; d < NDIM; ++d) {
      const float v = sm[CB_OFF + c * CB_PITCH + d];
      s += v * v;
    }
    sm[CN_OFF + c] = s;
  }
  __syncthreads();

  const int rowBase = wave * 16 + half * 8;   // + j = this lane's row for stat j

  // ---- pass 1: logits = 2hp*(z.c) - hp*||z||^2 - hp*||c||^2 via WMMA f32 ----
  // 4 independent accumulator chains per tile for WMMA ILP; lane-local running
  // max across tiles (single butterfly reduction afterwards).
  float m1[8];
#pragma unroll
  for (int j = 0; j < 8; ++j) m1[j] = -3.4e38f;

  for (int t = 0; t < NK / 16; ++t) {
    const int c0 = t * 16;
    v8f acc[4];
#pragma unroll
    for (int c = 0; c < 4; ++c) acc[c] = (v8f){0, 0, 0, 0, 0, 0, 0, 0};
#pragma unroll
    for (int kk = 0; kk < 16; ++kk) {
      // B(4x16) frag: lanes 0-15: k={0,1}, n=lane; lanes 16-31: k={2,3}
      const float* p = &sm[CB_OFF + (c0 + rlane) * CB_PITCH + kk * 4 + half * 2];
      v2f b; b.x = p[0]; b.y = p[1];
      acc[kk & 3] = wmma_f32(az[kk], b, acc[kk & 3]);
    }
    const v8f accT = (acc[0] + acc[1]) + (acc[2] + acc[3]);
    const float hcn = hp * sm[CN_OFF + c0 + rlane];
#pragma unroll
    for (int j = 0; j < 8; ++j) {
      const float lr = hp2 * accT[j] - hzn[j] - hcn;
      sm[LG_OFF + (rowBase + j) * NK + c0 + rlane] = lr;
      m1[j] = fmaxf(m1[j], lr);
    }
  }
  // one butterfly per row (within each 16-lane half)
#pragma unroll
  for (int j = 0; j < 8; ++j) {
    float tm = m1[j];
    tm = fmaxf(tm, __shfl_xor(tm, 1, 32));
    tm = fmaxf(tm, __shfl_xor(tm, 2, 32));
    tm = fmaxf(tm, __shfl_xor(tm, 4, 32));
    tm = fmaxf(tm, __shfl_xor(tm, 8, 32));
    m1[j] = tm;
  }

  // ---- pass 2a: s1 = sum e, t = sum e*l'; cache e over the logit slots ----
  float s1[8], tt[8];
#pragma unroll
  for (int j = 0; j < 8; ++j) { s1[j] = 0.0f; tt[j] = 0.0f; }
  for (int t = 0; t < NK / 16; ++t) {
#pragma unroll
    for (int j = 0; j < 8; ++j) {
      const int idx = LG_OFF + (rowBase + j) * NK + t * 16 + rlane;
      const float lr = sm[idx];
      const float lp = fminf(fmaxf(lr - m1[j], -50.0f), 50.0f);
      const float e  = __expf(lp);
      s1[j] += e;
      tt[j] += e * lp;
      sm[idx] = e;                       // reuse slot: e = exp(clipped logit)
    }
  }
  float invs1[8];
#pragma unroll
  for (int j = 0; j < 8; ++j) invs1[j] = 1.0f / s1[j];

  // ---- pass 2b: prob column sums + gumbel weights + GEMM2 (zq) ----
  // w = exp((l'+g)/T), T=0.5, g = -log(L), L = -log(u+eps)+eps  ==>  w = (e/L)^2
  v8f zqacc[4];
#pragma unroll
  for (int dt = 0; dt < 4; ++dt) zqacc[dt] = (v8f){0, 0, 0, 0, 0, 0, 0, 0};
  v8f oneacc = {0, 0, 0, 0, 0, 0, 0, 0};
  v2f bone; bone.x = 1.0f; bone.y = 1.0f;
  const size_t gubase = (size_t)(wg * ROWS_WG + rowBase) * NK + rlane;

  for (int t = 0; t < NK / 16; ++t) {
    const int c0 = t * 16;
    // prefetch next tile of the gumbel stream (dominant HBM traffic)
    if (t + 1 < NK / 16) {
#pragma unroll
      for (int j = 0; j < 8; ++j)
        __builtin_prefetch(&gu[gubase + (size_t)j * NK + c0 + 16], 0, 1);
    }
    float colp = 0.0f;
#pragma unroll
    for (int j = 0; j < 8; ++j) {
      const float e = sm[LG_OFF + (rowBase + j) * NK + c0 + rlane];
      colp += e * invs1[j];
      const float u = gu[gubase + (size_t)j * NK + c0];
      const float L = -__logf(u + 1e-10f) + 1e-10f;
      const float q = e * (1.0f / L);
      sm[WB_OFF + wave * 320 + (j + half * 8) * 20 + rlane] = q * q;
    }
    // per-code probability partial sums (16 rows of this wave)
    colp += __shfl_xor(colp, 16, 32);
    if (half == 0) atomicAdd(&sm[AP_OFF + c0 + rlane], colp);

    asm volatile("" ::: "memory");  // keep ds stores before the re-layout loads
    // re-read weights as A fragments (rows = lanes)
    v2f aw[4];
#pragma unroll
    for (int kk = 0; kk < 4; ++kk) {
      const float* p = &sm[WB_OFF + wave * 320 + rlane * 20 + kk * 4 + half * 2];
      aw[kk].x = p[0]; aw[kk].y = p[1];
    }
#pragma unroll
    for (int kk = 0; kk < 4; ++kk) {
      oneacc = wmma_f32(aw[kk], bone, oneacc);   // row sums of w (softmax denom)
#pragma unroll
      for (int dt = 0; dt < 4; ++dt) {
        // B(4x16): k = code (c0+kk*4 + half*2 + {0,1}), n = dim (dt*16 + lane)
        const float* p = &sm[CB_OFF + (c0 + kk * 4 + half * 2) * CB_PITCH + dt * 16 + rlane];
        v2f bc; bc.x = p[0]; bc.y = p[CB_PITCH];
        zqacc[dt] = wmma_f32(aw[kk], bc, zqacc[dt]);
      }
    }
  }

  // per-row sum p*log p = t/s1 - log s1
  float plpSum = 0.0f;
#pragma unroll
  for (int j = 0; j < 8; ++j) plpSum += tt[j] * invs1[j] - __logf(s1[j]);
  plpSum += __shfl_xor(plpSum, 16, 32);   // all lanes: total over wave's 16 rows

  // ---- pass 3: finalize zq, weighted squared error, stage transposed zq ----
  float inv2[8];
#pragma unroll
  for (int j = 0; j < 8; ++j) inv2[j] = 1.0f / oneacc[j];
  float errSum = 0.0f;
#pragma unroll
  for (int dt = 0; dt < 4; ++dt) {
#pragma unroll
    for (int j = 0; j < 8; ++j) {
      const float q = zqacc[dt][j] * inv2[j];
      const int d = dt * 16 + rlane;
      const int r = rowBase + j;
      const float df = sm[ZT_OFF + d * 68 + r] - q;
      errSum += df * df;
      sm[ZQ_OFF + d * 68 + r] = q;
    }
  }
  errSum += __shfl_xor(errSum, 1, 32);
  errSum += __shfl_xor(errSum, 2, 32);
  errSum += __shfl_xor(errSum, 4, 32);
  errSum += __shfl_xor(errSum, 8, 32);
  errSum += __shfl_xor(errSum, 16, 32);

  if (lane == 0) { sm[SC_OFF + wave] = plpSum; sm[SC_OFF + 8 + wave] = errSum; }
  __syncthreads();

  if (tid == 0) {
    const float a = sm[SC_OFF] + sm[SC_OFF + 1] + sm[SC_OFF + 2] + sm[SC_OFF + 3];
    const float b = sm[SC_OFF + 8] + sm[SC_OFF + 9] + sm[SC_OFF + 10] + sm[SC_OFF + 11];
    atomicAdd(&ws[0], a);
    atomicAdd(&ws[1], b);
  }
  for (int c = tid; c < NK; c += 128) atomicAdd(&ws[2 + c], sm[AP_OFF + c]);

  // ---- coalesced store of z_to_decoder [bs, D, W, H] ----
  const int n0 = wg * ROWS_WG;
  const size_t obase = (size_t)(n0 >> 10) * 65536 + (n0 & 1023);
  for (int it = 0; it < 32; ++it) {
    const int d = it * 2 + (tid >> 6);
    const int i = tid & 63;
    out[obase + (size_t)d * 1024 + i] = sm[ZQ_OFF + d * 68 + i];
  }
}

__global__ void gvq_init(float* __restrict__ ws) {
  const int t = blockIdx.x * blockDim.x + threadIdx.x;
  if (t < 2 + NK) ws[t] = 0.0f;
}

__global__ void gvq_final(const float* __restrict__ var_q,
                          const float* __restrict__ ws, float* __restrict__ out) {
  __shared__ float red[NK];
  const int t = threadIdx.x;
  float a = ws[2 + t] * (1.0f / (float)NTOT);     // avg prob per code
  a = fmaxf(a, 1e-10f);
  red[t] = a * __logf(a + 1e-10f);
  __syncthreads();
  for (int s = NK / 2; s > 0; s >>= 1) {
    if (t < s) red[t] += red[t + s];
    __syncthreads();
  }
  if (t == 0) {
    const float vq = fmaxf(var_q[0], 1e-5f);
    const float hp = 0.5f * fminf(1.0f / vq, 1e5f);
    const float loss = ws[0] / (float)NBS + ws[1] * hp / (float)NBS;
    out[NTOT * NDIM]     = loss;
    out[NTOT * NDIM + 1] = __expf(-red[0]);       // perplexity
  }
}

extern "C" void kernel_launch(void* const* d_in, const int* in_sizes, int n_in,
                              void* d_out, int out_size, void* d_ws, size_t ws_size,
                              hipStream_t stream) {
  const float* z   = (const float*)d_in[0];
  const float* vq  = (const float*)d_in[1];
  const float* cb  = (const float*)d_in[2];
  const float* gu  = (const float*)d_in[3];
  float* out = (float*)d_out;
  float* ws  = (float*)d_ws;

  gvq_init<<<(2 + NK + 255) / 256, 256, 0, stream>>>(ws);
  gvq_main<<<NWG, 128, LDS_FLOATS * sizeof(float), stream>>>(z, vq, cb, gu, out, ws);
  gvq_final<<<1, NK, 0, stream>>>(vq, ws, out);
}